// MultiDilatelocalAttention3D_67473936220610
// MI455X (gfx1250) — compile-verified
//
#include <hip/hip_runtime.h>

typedef __attribute__((ext_vector_type(2))) float v2f;
typedef __attribute__((ext_vector_type(8))) float v8f;

#define NPOS 9216
#define ZD 16
#define YD 24
#define XD 24
#define CDIM 192
#define QKVDIM 576

// ---------------------------------------------------------------------------
// C(MxN) = A(MxK, row-major) * B^T   (B stored NxK row-major)   [+ bias]
// One wave computes a 32(M) x 64(N) tile with v_wmma_f32_16x16x4_f32:
// 8 accumulators (2 M-halves x 4 N-tiles), software-pipelined fragment loads.
//
// f32 WMMA operand layout (CDNA5 ISA 7.12.2):
//   A 16x4 : lane l holds A[l%16, 2*(l/16)+j] in vgpr j (j=0,1)
//   B 4x16 : lane l holds B[2*(l/16)+j, l%16] in vgpr j
//   C/D    : vgpr r, lanes 0-15 -> row r, lanes 16-31 -> row r+8; col = l%16
// ---------------------------------------------------------------------------
template <int K>
__global__ __launch_bounds__(256) void wmma_gemm_nt(
    const float* __restrict__ A, const float* __restrict__ B,
    const float* __restrict__ bias, float* __restrict__ C,
    int M, int N, int nGroups) {
  int wave   = (int)((blockIdx.x * blockDim.x + threadIdx.x) >> 5);
  int lane   = (int)(threadIdx.x & 31);
  int mTile  = wave / nGroups;
  int nGroup = wave - mTile * nGroups;
  int m0 = mTile << 5;   // 32 rows of M per wave
  int n0 = nGroup << 6;  // 64 cols of N per wave
  if (m0 >= M) return;   // wave-uniform

  int r16   = lane & 15;
  int khalf = (lane >> 4) << 1;  // 0 or 2

  // Single base pointers; the 16*K-float tile strides are compile-time
  // constants -> folded into 24-bit instruction offsets.
  const float* aptr = A + (size_t)(m0 + r16) * K + khalf;
  const float* bptr = B + (size_t)(n0 + r16) * K + khalf;

  v8f c00 = {}, c01 = {}, c02 = {}, c03 = {};
  v8f c10 = {}, c11 = {}, c12 = {}, c13 = {};

  // Prologue: fragments for k0 = 0.
  v2f a0 = *(const v2f*)(aptr);
  v2f a1 = *(const v2f*)(aptr + 16 * K);
  v2f b0 = *(const v2f*)(bptr);
  v2f b1 = *(const v2f*)(bptr + 16 * K);
  v2f b2 = *(const v2f*)(bptr + 32 * K);
  v2f b3 = *(const v2f*)(bptr + 48 * K);

#pragma unroll 4
  for (int k0 = 0; k0 < K - 4; k0 += 4) {
    // Issue next-step loads BEFORE consuming current fragments so the
    // WMMAs below only need s_wait_loadcnt <= outstanding-next-loads.
    v2f an0 = *(const v2f*)(aptr + k0 + 4);
    v2f an1 = *(const v2f*)(aptr + k0 + 4 + 16 * K);
    v2f bn0 = *(const v2f*)(bptr + k0 + 4);
    v2f bn1 = *(const v2f*)(bptr + k0 + 4 + 16 * K);
    v2f bn2 = *(const v2f*)(bptr + k0 + 4 + 32 * K);
    v2f bn3 = *(const v2f*)(bptr + k0 + 4 + 48 * K);

    c00 = __builtin_amdgcn_wmma_f32_16x16x4_f32(false, a0, false, b0, (short)0, c00, false, false);
    c01 = __builtin_amdgcn_wmma_f32_16x16x4_f32(false, a0, false, b1, (short)0, c01, false, false);
    c02 = __builtin_amdgcn_wmma_f32_16x16x4_f32(false, a0, false, b2, (short)0, c02, false, false);
    c03 = __builtin_amdgcn_wmma_f32_16x16x4_f32(false, a0, false, b3, (short)0, c03, false, false);
    c10 = __builtin_amdgcn_wmma_f32_16x16x4_f32(false, a1, false, b0, (short)0, c10, false, false);
    c11 = __builtin_amdgcn_wmma_f32_16x16x4_f32(false, a1, false, b1, (short)0, c11, false, false);
    c12 = __builtin_amdgcn_wmma_f32_16x16x4_f32(false, a1, false, b2, (short)0, c12, false, false);
    c13 = __builtin_amdgcn_wmma_f32_16x16x4_f32(false, a1, false, b3, (short)0, c13, false, false);

    a0 = an0; a1 = an1;
    b0 = bn0; b1 = bn1; b2 = bn2; b3 = bn3;
  }
  // Epilogue: last K-step.
  c00 = __builtin_amdgcn_wmma_f32_16x16x4_f32(false, a0, false, b0, (short)0, c00, false, false);
  c01 = __builtin_amdgcn_wmma_f32_16x16x4_f32(false, a0, false, b1, (short)0, c01, false, false);
  c02 = __builtin_amdgcn_wmma_f32_16x16x4_f32(false, a0, false, b2, (short)0, c02, false, false);
  c03 = __builtin_amdgcn_wmma_f32_16x16x4_f32(false, a0, false, b3, (short)0, c03, false, false);
  c10 = __builtin_amdgcn_wmma_f32_16x16x4_f32(false, a1, false, b0, (short)0, c10, false, false);
  c11 = __builtin_amdgcn_wmma_f32_16x16x4_f32(false, a1, false, b1, (short)0, c11, false, false);
  c12 = __builtin_amdgcn_wmma_f32_16x16x4_f32(false, a1, false, b2, (short)0, c12, false, false);
  c13 = __builtin_amdgcn_wmma_f32_16x16x4_f32(false, a1, false, b3, (short)0, c13, false, false);

  int col = lane & 15;
  float bias0 = 0.f, bias1 = 0.f, bias2 = 0.f, bias3 = 0.f;
  if (bias) {
    bias0 = bias[n0 + col];
    bias1 = bias[n0 + 16 + col];
    bias2 = bias[n0 + 32 + col];
    bias3 = bias[n0 + 48 + col];
  }
  int rbase0 = m0 + ((lane >> 4) << 3);        // M half 0
  int rbase1 = m0 + 16 + ((lane >> 4) << 3);   // M half 1
#pragma unroll
  for (int r = 0; r < 8; ++r) {
    float* crow0 = C + (size_t)(rbase0 + r) * N + n0 + col;
    crow0[0]  = c00[r] + bias0;
    crow0[16] = c01[r] + bias1;
    crow0[32] = c02[r] + bias2;
    crow0[48] = c03[r] + bias3;
    float* crow1 = C + (size_t)(rbase1 + r) * N + n0 + col;
    crow1[0]  = c10[r] + bias0;
    crow1[16] = c11[r] + bias1;
    crow1[32] = c12[r] + bias2;
    crow1[48] = c13[r] + bias3;
  }
}

// ---------------------------------------------------------------------------
// Multi-dilation 3x3x3 local attention. One lane = one (position, dil, head).
// qkv rows: [q(192) | k(192) | v(192)] per point; branch i uses channels
// i*64 + head*32 .. +32 of each section. Zero padding => OOB neighbor has
// logit 0 (participates in softmax) and v = 0.
// ---------------------------------------------------------------------------
__global__ __launch_bounds__(256) void dilate_attn(const float* __restrict__ qkv,
                                                   float* __restrict__ yout) {
  int t    = (int)(blockIdx.x * blockDim.x + threadIdx.x);
  int pos  = t % NPOS;
  int comb = t / NPOS;  // 0..5
  if (comb >= 6) return;
  int dilIdx = comb >> 1;
  int head   = comb & 1;
  int dil    = dilIdx + 1;
  int xi = pos % XD;
  int yi = (pos / XD) % YD;
  int zi = pos / (XD * YD);
  int chOff = dilIdx * 64 + head * 32;

  const float4* qp = (const float4*)(qkv + (size_t)pos * QKVDIM + chOff);
  float4 q[8];
#pragma unroll
  for (int j = 0; j < 8; ++j) q[j] = qp[j];

  const float scale = 0.1767766952966369f;  // 1/sqrt(32)
  float logits[27];
  float mx = -3.4e38f;
#pragma unroll
  for (int n = 0; n < 27; ++n) {
    int dz = n / 9 - 1, dy = (n / 3) % 3 - 1, dx = n % 3 - 1;
    int nz = zi + dz * dil, ny = yi + dy * dil, nx = xi + dx * dil;
    float lg = 0.0f;
    if (nz >= 0 && nz < ZD && ny >= 0 && ny < YD && nx >= 0 && nx < XD) {
      int np = (nz * YD + ny) * XD + nx;
      const float4* kp = (const float4*)(qkv + (size_t)np * QKVDIM + CDIM + chOff);
      float d = 0.0f;
#pragma unroll
      for (int j = 0; j < 8; ++j) {
        float4 kk = kp[j];
        d += q[j].x * kk.x + q[j].y * kk.y + q[j].z * kk.z + q[j].w * kk.w;
      }
      lg = d * scale;
    }
    logits[n] = lg;
    mx = fmaxf(mx, lg);
  }

  float4 acc[8];
#pragma unroll
  for (int j = 0; j < 8; ++j) acc[j] = make_float4(0.f, 0.f, 0.f, 0.f);
  float wsum = 0.0f;
#pragma unroll
  for (int n = 0; n < 27; ++n) {
    int dz = n / 9 - 1, dy = (n / 3) % 3 - 1, dx = n % 3 - 1;
    int nz = zi + dz * dil, ny = yi + dy * dil, nx = xi + dx * dil;
    float w = __expf(logits[n] - mx);
    wsum += w;
    if (nz >= 0 && nz < ZD && ny >= 0 && ny < YD && nx >= 0 && nx < XD) {
      int np = (nz * YD + ny) * XD + nx;
      const float4* vp = (const float4*)(qkv + (size_t)np * QKVDIM + 2 * CDIM + chOff);
#pragma unroll
      for (int j = 0; j < 8; ++j) {
        float4 vv = vp[j];
        acc[j].x += w * vv.x; acc[j].y += w * vv.y;
        acc[j].z += w * vv.z; acc[j].w += w * vv.w;
      }
    }
  }
  float inv = 1.0f / wsum;
  float4* op = (float4*)(yout + (size_t)pos * CDIM + chOff);
#pragma unroll
  for (int j = 0; j < 8; ++j)
    op[j] = make_float4(acc[j].x * inv, acc[j].y * inv, acc[j].z * inv, acc[j].w * inv);
}

// ---------------------------------------------------------------------------
extern "C" void kernel_launch(void* const* d_in, const int* in_sizes, int n_in,
                              void* d_out, int out_size, void* d_ws, size_t ws_size,
                              hipStream_t stream) {
  const float* x      = (const float*)d_in[0];  // (1,16,24,24,192)
  const float* qkv_w  = (const float*)d_in[1];  // (576,192)
  const float* proj_w = (const float*)d_in[2];  // (192,192)
  const float* proj_b = (const float*)d_in[3];  // (192,)

  float* qkv  = (float*)d_ws;                   // 9216 x 576
  float* ybuf = qkv + (size_t)NPOS * QKVDIM;    // 9216 x 192

  // 1) QKV projection: (9216x192) x (576x192)^T -> (9216x576)
  {
    int waves   = (NPOS / 32) * (QKVDIM / 64);  // 288 * 9 = 2592
    int threads = waves * 32;                   // 82944 = 324 * 256
    wmma_gemm_nt<CDIM><<<threads / 256, 256, 0, stream>>>(
        x, qkv_w, nullptr, qkv, NPOS, QKVDIM, QKVDIM / 64);
  }

  // 2) Multi-dilation local attention -> (9216x192)
  {
    int tasks = 6 * NPOS;  // 55296 = 216 * 256
    dilate_attn<<<tasks / 256, 256, 0, stream>>>(qkv, ybuf);
  }

  // 3) Output projection + bias: (9216x192) x (192x192)^T -> (9216x192)
  {
    int waves   = (NPOS / 32) * (CDIM / 64);  // 288 * 3 = 864
    int threads = waves * 32;                 // 27648 = 108 * 256
    wmma_gemm_nt<CDIM><<<threads / 256, 256, 0, stream>>>(
        ybuf, proj_w, proj_b, (float*)d_out, NPOS, CDIM, CDIM / 64);
  }
}